// GcnNet_90288802496747
// MI455X (gfx1250) — compile-verified
//
#include <hip/hip_runtime.h>

// ---------------------------------------------------------------------------
// GCN (3x GCNConv + mean-pool + MLP) for MI455X (gfx1250, wave32, WMMA).
// Dense GEMMs use V_WMMA_F32_16X16X4_F32 (full fp32). Sparse aggregation uses
// coalesced float4 gathers + global fp32 atomics (bandwidth-bound path).
// ---------------------------------------------------------------------------

typedef __attribute__((ext_vector_type(2))) float v2f;
typedef __attribute__((ext_vector_type(8))) float v8f;

#define NHID 64
#define NMID 32
#define NCLS 10

// ---------------- degree / normalization ----------------------------------
__global__ void k_deg_init(float* __restrict__ deg, int n) {
  int i = blockIdx.x * blockDim.x + threadIdx.x;
  if (i < n) deg[i] = 1.0f;  // self-loop contribution
}

__global__ void k_deg_edges(const int* __restrict__ col, float* __restrict__ deg, int e) {
  int i = blockIdx.x * blockDim.x + threadIdx.x;
  if (i < e) atomicAdd(&deg[col[i]], 1.0f);
}

__global__ void k_dinv(float* __restrict__ deg, int n) {
  int i = blockIdx.x * blockDim.x + threadIdx.x;
  if (i < n) deg[i] = rsqrtf(deg[i]);  // deg >= 1 always (self-loops)
}

// ---------------- layer 1: X(N x 2) @ W1(2 x 64) ---------------------------
__global__ void k_lin2(const float* __restrict__ x, const float* __restrict__ W,
                       float* __restrict__ H, int n) {
  long long gid = (long long)blockIdx.x * blockDim.x + threadIdx.x;
  if (gid >= (long long)n * NHID) return;
  int i = (int)(gid >> 6);
  int c = (int)(gid & 63);
  H[gid] = fmaf(x[2 * i], W[c], x[2 * i + 1] * W[NHID + c]);
}

// ---------------- WMMA GEMM: A(rows x 64) @ W(64 x OUTC) -------------------
// One wave computes a 16-row strip. fp32 WMMA 16x16x4, K-loop of 16.
// A frag (16x4 f32): lane(m,hi): v[j] = A[m][k0 + hi*2 + j]
// B frag (4x16 f32): lane(n,hi): v[j] = W[k0 + hi*2 + j][n]
// D frag (16x16 f32): lane(n,hi): v[j] = D[hi*8 + j][n]
template <int OUTC, bool FUSE_BIAS_RELU>
__global__ __launch_bounds__(256) void k_gemm_wmma(const float* __restrict__ A,
                                                   const float* __restrict__ W,
                                                   const float* __restrict__ bias,
                                                   float* __restrict__ H, int nrows) {
  int wave = blockIdx.x * (blockDim.x >> 5) + (threadIdx.x >> 5);
  int ntiles = nrows >> 4;
  if (wave >= ntiles) return;  // wave-uniform: EXEC stays all-ones for WMMA

  int lane = threadIdx.x & 31;
  int l15 = lane & 15;   // M index for A, N index for B/D
  int hi = lane >> 4;    // half-wave select
  int row0 = wave << 4;

  const float* arow = A + (size_t)(row0 + l15) * NHID;

#pragma unroll
  for (int n0 = 0; n0 < OUTC; n0 += 16) {
    v8f acc = {};
#pragma unroll
    for (int k0 = 0; k0 < NHID; k0 += 4) {
      int kk = k0 + hi * 2;
      v2f av, bv;
      av[0] = arow[kk];
      av[1] = arow[kk + 1];
      bv[0] = W[(size_t)kk * OUTC + n0 + l15];
      bv[1] = W[(size_t)(kk + 1) * OUTC + n0 + l15];
      acc = __builtin_amdgcn_wmma_f32_16x16x4_f32(false, av, false, bv,
                                                  (short)0, acc, false, false);
    }
#pragma unroll
    for (int j = 0; j < 8; ++j) {
      int mm = hi * 8 + j;
      float v = acc[j];
      if (FUSE_BIAS_RELU) v = fmaxf(v + bias[n0 + l15], 0.0f);
      H[(size_t)(row0 + mm) * OUTC + n0 + l15] = v;
    }
  }
}

// ---------------- edge scatter: AGG[col] += dinv[r]*dinv[c] * H[row] -------
__global__ void k_scatter(const int* __restrict__ row, const int* __restrict__ col,
                          const float* __restrict__ dinv, const float* __restrict__ H,
                          float* __restrict__ AGG, int e) {
  long long gid = (long long)blockIdx.x * blockDim.x + threadIdx.x;
  int ed = (int)(gid >> 4);
  if (ed >= e) return;
  int q = ((int)gid & 15) * 4;
  int r = row[ed];
  int c = col[ed];
  float w = dinv[r] * dinv[c];
  const float4 h = *(const float4*)(H + (size_t)r * NHID + q);
  float* o = AGG + (size_t)c * NHID + q;
  atomicAdd(o + 0, w * h.x);
  atomicAdd(o + 1, w * h.y);
  atomicAdd(o + 2, w * h.z);
  atomicAdd(o + 3, w * h.w);
}

// ---------------- finalize: relu(AGG + dinv^2 * H + b) ---------------------
__global__ void k_finalize(const float* __restrict__ AGG, const float* __restrict__ H,
                           const float* __restrict__ dinv, const float* __restrict__ b,
                           float* __restrict__ out, int n) {
  long long gid = (long long)blockIdx.x * blockDim.x + threadIdx.x;
  if (gid >= (long long)n * NHID) return;
  int i = (int)(gid >> 6);
  int c = (int)(gid & 63);
  float di = dinv[i];
  float v = AGG[gid] + di * di * H[gid] + b[c];
  out[gid] = fmaxf(v, 0.0f);
}

// ---------------- global mean pool -----------------------------------------
__global__ void k_pool_acc(const float* __restrict__ h, const int* __restrict__ batch,
                           float* __restrict__ psum, float* __restrict__ pcnt, int n) {
  long long gid = (long long)blockIdx.x * blockDim.x + threadIdx.x;
  if (gid >= (long long)n * NHID) return;
  int i = (int)(gid >> 6);
  int c = (int)(gid & 63);
  int g = batch[i];
  atomicAdd(&psum[(size_t)g * NHID + c], h[gid]);
  if (c == 0) atomicAdd(&pcnt[g], 1.0f);
}

__global__ void k_pool_div(float* __restrict__ psum, const float* __restrict__ pcnt, int g) {
  int gid = blockIdx.x * blockDim.x + threadIdx.x;
  if (gid >= g * NHID) return;
  psum[gid] /= fmaxf(pcnt[gid >> 6], 1.0f);
}

// ---------------- head: z(32) @ Wf2(32x10) + bf2, log_softmax --------------
__global__ void k_head(const float* __restrict__ z, const float* __restrict__ Wf2,
                       const float* __restrict__ bf2, float* __restrict__ out, int g) {
  int i = blockIdx.x * blockDim.x + threadIdx.x;
  if (i >= g) return;
  const float* zi = z + (size_t)i * NMID;
  float logit[NCLS];
#pragma unroll
  for (int o = 0; o < NCLS; ++o) {
    float s = bf2[o];
#pragma unroll
    for (int k = 0; k < NMID; ++k) s = fmaf(zi[k], Wf2[k * NCLS + o], s);
    logit[o] = s;
  }
  float mx = logit[0];
#pragma unroll
  for (int o = 1; o < NCLS; ++o) mx = fmaxf(mx, logit[o]);
  float se = 0.0f;
#pragma unroll
  for (int o = 0; o < NCLS; ++o) se += expf(logit[o] - mx);
  float lse = mx + logf(se);
#pragma unroll
  for (int o = 0; o < NCLS; ++o) out[(size_t)i * NCLS + o] = logit[o] - lse;
}

// ---------------------------------------------------------------------------
extern "C" void kernel_launch(void* const* d_in, const int* in_sizes, int n_in,
                              void* d_out, int out_size, void* d_ws, size_t ws_size,
                              hipStream_t stream) {
  const float* x = (const float*)d_in[0];
  const int* ei = (const int*)d_in[1];    // (2, E) flat; JAX x64 off -> int32
  const int* batch = (const int*)d_in[2];
  const float* W1 = (const float*)d_in[3];
  const float* b1 = (const float*)d_in[4];
  const float* W2 = (const float*)d_in[5];
  const float* b2 = (const float*)d_in[6];
  const float* W3 = (const float*)d_in[7];
  const float* b3 = (const float*)d_in[8];
  const float* Wf1 = (const float*)d_in[9];
  const float* bf1 = (const float*)d_in[10];
  const float* Wf2 = (const float*)d_in[11];
  const float* bf2 = (const float*)d_in[12];

  const int n = in_sizes[0] / 2;   // nodes
  const int e = in_sizes[1] / 2;   // edges
  const int g = out_size / NCLS;   // graphs
  const int* row = ei;             // sources (gather)
  const int* col = ei + e;         // targets (scatter)

  // workspace partition (fp32)
  float* ws = (float*)d_ws;
  float* dinv = ws;                              // n           (n % 64 == 0)
  float* bufA = dinv + n;                        // n*64
  float* bufB = bufA + (size_t)n * NHID;         // n*64
  float* psum = bufB + (size_t)n * NHID;         // g*64
  float* pcnt = psum + (size_t)g * NHID;         // g
  float* zbuf = pcnt + g;                        // g*32
  (void)ws_size; (void)n_in;

  const int tpb = 256;
  const int nb_n = (n + tpb - 1) / tpb;
  const int nb_e = (e + tpb - 1) / tpb;
  const long long nf = (long long)n * NHID;
  const int nb_nf = (int)((nf + tpb - 1) / tpb);
  const int nb_ef = (int)(((long long)e * 16 + tpb - 1) / tpb);
  const int nb_gemm = ((n >> 4) + 7) / 8;     // 8 waves per block

  // symmetric GCN normalization: dinv = rsqrt(deg + self-loop)
  k_deg_init<<<nb_n, tpb, 0, stream>>>(dinv, n);
  k_deg_edges<<<nb_e, tpb, 0, stream>>>(col, dinv, e);
  k_dinv<<<nb_n, tpb, 0, stream>>>(dinv, n);

  // ---- layer 1 ----
  k_lin2<<<nb_nf, tpb, 0, stream>>>(x, W1, bufA, n);
  hipMemsetAsync(bufB, 0, (size_t)nf * sizeof(float), stream);
  k_scatter<<<nb_ef, tpb, 0, stream>>>(row, col, dinv, bufA, bufB, e);
  k_finalize<<<nb_nf, tpb, 0, stream>>>(bufB, bufA, dinv, b1, bufA, n);

  // ---- layer 2 ----
  k_gemm_wmma<NHID, false><<<nb_gemm, tpb, 0, stream>>>(bufA, W2, nullptr, bufB, n);
  hipMemsetAsync(bufA, 0, (size_t)nf * sizeof(float), stream);
  k_scatter<<<nb_ef, tpb, 0, stream>>>(row, col, dinv, bufB, bufA, e);
  k_finalize<<<nb_nf, tpb, 0, stream>>>(bufA, bufB, dinv, b2, bufA, n);

  // ---- layer 3 ----
  k_gemm_wmma<NHID, false><<<nb_gemm, tpb, 0, stream>>>(bufA, W3, nullptr, bufB, n);
  hipMemsetAsync(bufA, 0, (size_t)nf * sizeof(float), stream);
  k_scatter<<<nb_ef, tpb, 0, stream>>>(row, col, dinv, bufB, bufA, e);
  k_finalize<<<nb_nf, tpb, 0, stream>>>(bufA, bufB, dinv, b3, bufA, n);

  // ---- global mean pool ----
  hipMemsetAsync(psum, 0, (size_t)(g * NHID + g) * sizeof(float), stream);
  k_pool_acc<<<nb_nf, tpb, 0, stream>>>(bufA, batch, psum, pcnt, n);
  k_pool_div<<<(g * NHID + tpb - 1) / tpb, tpb, 0, stream>>>(psum, pcnt, g);

  // ---- MLP head ----
  k_gemm_wmma<NMID, true><<<((g >> 4) + 7) / 8, tpb, 0, stream>>>(psum, Wf1, bf1, zbuf, g);
  k_head<<<(g + tpb - 1) / tpb, tpb, 0, stream>>>(zbuf, Wf2, bf2, (float*)d_out, g);
}